// VGAE_369367187668
// MI455X (gfx1250) — compile-verified
//
#include <hip/hip_runtime.h>
#include <hip/hip_bf16.h>

// MI455X / gfx1250 VGAE forward.
// FP32 WMMA (V_WMMA_F32_16X16X4_F32) for all GEMMs: inputs are fp32 and the
// whole working set (<50 MB) lives in the 192 MB L2, so GEMM operand reuse is
// serviced by L2. Each wave register-tiles a 32x32 C block (2x2 WMMA tiles)
// so A/B fragments are reused 2x each, and the K-loop is explicitly software-
// pipelined (next K-step's fragments loaded before current step's WMMAs) so
// global-load latency hides behind the matrix pipe instead of draining
// loadcnt before every WMMA pair.
// The decoder uses LDS tiles since each P/Q row is reused 16x inside a tile.

typedef __attribute__((ext_vector_type(2))) float v2f;
typedef __attribute__((ext_vector_type(8))) float v8f;

#define NEG_SLOPE 0.01f

// ---------------------------------------------------------------------------
// Generic fp32 WMMA GEMM: C[M,N] = op(A[M,K] @ B)  (+ fused epilogue)
//   BT=false: B is [K,N] row-major (access B[k*ldb + n])
//   BT=true : B is [N,K] row-major (access B[n*ldb + k])  -> computes A @ B^T
//   EPI=0: none   EPI=1: bias add + leaky_relu
// Block = 128 threads = 4 waves in a 2x2 arrangement -> 64x64 block tile.
// Each wave: 32x32 output block = 2x2 accumulators of 16x16.
// Requires K % 4 == 0 and K >= 8 (true for all call sites here).
// WMMA fp32 fragment layouts per CDNA5 ISA 7.12.2:
//   A (16x4):  lane l -> row = l%16, VGPR v -> K = 2*(l/16) + v
//   B (4x16):  lane l -> col = l%16, VGPR v -> K = 2*(l/16) + v
//   D (16x16): lane l -> col = l%16, VGPR r -> row = r + 8*(l/16)
// ---------------------------------------------------------------------------
template <int EPI, bool BT>
__global__ __launch_bounds__(128) void gemm_wmma_f32(
    const float* __restrict__ A, const float* __restrict__ B,
    const float* __restrict__ bias, float* __restrict__ C,
    int M, int N, int K, int lda, int ldb, int ldc)
{
    const int wave  = threadIdx.x >> 5;             // 0..3
    const int lane  = threadIdx.x & 31;
    const int l16   = lane & 15;
    const int hi    = lane >> 4;                    // 0 or 1
    const int waveN = wave & 1;
    const int waveM = wave >> 1;

    const int tileM = blockIdx.y * 64 + waveM * 32;
    const int tileN = blockIdx.x * 64 + waveN * 32;

    const int aRow0 = tileM + l16;
    const int col0  = tileN + l16;
    const int col1  = col0 + 16;

    // Per-lane base pointers; in-loop address math is pure pointer increment.
    const float* pA0 = A + (size_t)aRow0 * lda + hi * 2;
    const float* pA1 = pA0 + (size_t)16 * lda;
    const float* pB0;
    const float* pB1;
    if (BT) {
        pB0 = B + (size_t)col0 * ldb + hi * 2;
        pB1 = B + (size_t)col1 * ldb + hi * 2;
    } else {
        pB0 = B + (size_t)(hi * 2) * ldb + col0;
        pB1 = pB0 + 16;
    }
    const size_t bStep = BT ? 4 : (size_t)4 * ldb;

    v8f acc00 = {0.f, 0.f, 0.f, 0.f, 0.f, 0.f, 0.f, 0.f};
    v8f acc01 = acc00, acc10 = acc00, acc11 = acc00;

    // ---- software pipeline: prologue loads K-step 0 ----
    v2f a0 = *(const v2f*)pA0;
    v2f a1 = *(const v2f*)pA1;
    v2f b0, b1;
    if (BT) {
        b0 = *(const v2f*)pB0;
        b1 = *(const v2f*)pB1;
    } else {
        b0.x = pB0[0]; b0.y = pB0[ldb];
        b1.x = pB1[0]; b1.y = pB1[ldb];
    }
    pA0 += 4; pA1 += 4; pB0 += bStep; pB1 += bStep;

    #pragma unroll 4
    for (int k0 = 4; k0 < K; k0 += 4) {
        // loads for NEXT K-step issue before current step's WMMAs
        v2f na0 = *(const v2f*)pA0;
        v2f na1 = *(const v2f*)pA1;
        v2f nb0, nb1;
        if (BT) {
            nb0 = *(const v2f*)pB0;
            nb1 = *(const v2f*)pB1;
        } else {
            nb0.x = pB0[0]; nb0.y = pB0[ldb];
            nb1.x = pB1[0]; nb1.y = pB1[ldb];
        }
        pA0 += 4; pA1 += 4; pB0 += bStep; pB1 += bStep;

        acc00 = __builtin_amdgcn_wmma_f32_16x16x4_f32(false, a0, false, b0, (short)0, acc00, false, false);
        acc01 = __builtin_amdgcn_wmma_f32_16x16x4_f32(false, a0, false, b1, (short)0, acc01, false, false);
        acc10 = __builtin_amdgcn_wmma_f32_16x16x4_f32(false, a1, false, b0, (short)0, acc10, false, false);
        acc11 = __builtin_amdgcn_wmma_f32_16x16x4_f32(false, a1, false, b1, (short)0, acc11, false, false);

        a0 = na0; a1 = na1; b0 = nb0; b1 = nb1;
    }

    // ---- pipeline epilogue: last K-step ----
    acc00 = __builtin_amdgcn_wmma_f32_16x16x4_f32(false, a0, false, b0, (short)0, acc00, false, false);
    acc01 = __builtin_amdgcn_wmma_f32_16x16x4_f32(false, a0, false, b1, (short)0, acc01, false, false);
    acc10 = __builtin_amdgcn_wmma_f32_16x16x4_f32(false, a1, false, b0, (short)0, acc10, false, false);
    acc11 = __builtin_amdgcn_wmma_f32_16x16x4_f32(false, a1, false, b1, (short)0, acc11, false, false);

    float bv0 = 0.f, bv1 = 0.f;
    if (EPI == 1) { bv0 = bias[col0]; bv1 = bias[col1]; }

    #pragma unroll
    for (int r = 0; r < 8; ++r) {
        const int row0 = tileM + r + hi * 8;
        const int row1 = row0 + 16;
        float v00 = acc00[r], v01 = acc01[r], v10 = acc10[r], v11 = acc11[r];
        if (EPI == 1) {
            v00 += bv0; v01 += bv1; v10 += bv0; v11 += bv1;
            v00 = (v00 >= 0.f) ? v00 : NEG_SLOPE * v00;
            v01 = (v01 >= 0.f) ? v01 : NEG_SLOPE * v01;
            v10 = (v10 >= 0.f) ? v10 : NEG_SLOPE * v10;
            v11 = (v11 >= 0.f) ? v11 : NEG_SLOPE * v11;
        }
        C[(size_t)row0 * ldc + col0] = v00;
        C[(size_t)row0 * ldc + col1] = v01;
        C[(size_t)row1 * ldc + col0] = v10;
        C[(size_t)row1 * ldc + col1] = v11;
    }
}

// ---------------------------------------------------------------------------
// Per-row fused tail of the encoder: mu, logvar (written straight to d_out),
// Z = mu + eps*exp(0.5*logvar), then P = Z @ Wd1[:, :L]^T, Q = Z @ Wd1[:, L:]^T.
// One 64-thread block per graph node (row). All tiny K, VALU-bound.
// ---------------------------------------------------------------------------
__global__ __launch_bounds__(64) void encode_tail(
    const float* __restrict__ AH, const float* __restrict__ eps,
    const float* __restrict__ Wmu, const float* __restrict__ bmu,
    const float* __restrict__ Wlv, const float* __restrict__ blv,
    const float* __restrict__ Wd1,
    float* __restrict__ mu_out, float* __restrict__ lv_out,
    float* __restrict__ P, float* __restrict__ Q,
    int Hdim, int L)
{
    __shared__ float s_mu[16];
    __shared__ float s_lv[16];
    __shared__ float s_z[16];

    const int i = blockIdx.x;
    const int t = threadIdx.x;
    const float* __restrict__ ah = AH + (size_t)i * Hdim;

    if (t < 32) {
        const int l = t & 15;
        const float* __restrict__ W = (t < 16) ? (Wmu + l * Hdim) : (Wlv + l * Hdim);
        float acc = (t < 16) ? bmu[l] : blv[l];
        #pragma unroll 8
        for (int h = 0; h < Hdim; ++h) acc = fmaf(ah[h], W[h], acc);
        if (t < 16) { s_mu[l] = acc; mu_out[i * L + l] = acc; }
        else        { s_lv[l] = acc; lv_out[i * L + l] = acc; }
    }
    __syncthreads();
    if (t < 16) {
        s_z[t] = s_mu[t] + eps[i * L + t] * expf(0.5f * s_lv[t]);
    }
    __syncthreads();

    // P[i,h] = sum_l Z[l] * Wd1[h, l];  Q[i,h] = sum_l Z[l] * Wd1[h, L+l]
    const int h = t & 31;
    const float* __restrict__ wrow = Wd1 + h * 32 + ((t < 32) ? 0 : 16);
    float acc = 0.f;
    #pragma unroll
    for (int l = 0; l < 16; ++l) acc = fmaf(s_z[l], wrow[l], acc);
    if (t < 32) P[i * 32 + h] = acc;
    else        Q[i * 32 + h] = acc;
}

// ---------------------------------------------------------------------------
// Decoder: A_pred[i,j] = sigmoid( sum_h leaky(P[i,h]+Q[j,h]+bd1[h]) * Wd2[h] + bd2 )
// 16x16 output tile per block; P/Q rows staged in LDS (16x reuse each),
// fully unrolled 32-wide leaky-FMA chain per output element.
// ---------------------------------------------------------------------------
__global__ __launch_bounds__(256) void decode_tile(
    const float* __restrict__ P, const float* __restrict__ Q,
    const float* __restrict__ bd1, const float* __restrict__ Wd2,
    const float* __restrict__ bd2, float* __restrict__ Apred, int N)
{
    __shared__ float sP[16][32];
    __shared__ float sQ[16][32];
    __shared__ float sB[32];
    __shared__ float sW[32];

    const int t  = threadIdx.x;
    const int i0 = blockIdx.y * 16;
    const int j0 = blockIdx.x * 16;

    {   // 512 floats per tile; 256 threads load float2 each
        const int r  = t >> 4;
        const int c2 = (t & 15) * 2;
        *(float2*)&sP[r][c2] = *(const float2*)&P[(size_t)(i0 + r) * 32 + c2];
        *(float2*)&sQ[r][c2] = *(const float2*)&Q[(size_t)(j0 + r) * 32 + c2];
    }
    if (t < 32) { sB[t] = bd1[t]; sW[t] = Wd2[t]; }
    __syncthreads();

    const int tj = t & 15;
    const int ti = t >> 4;
    float acc = 0.f;
    #pragma unroll
    for (int h = 0; h < 32; ++h) {
        float v = sP[ti][h] + sQ[tj][h] + sB[h];
        v = (v >= 0.f) ? v : NEG_SLOPE * v;
        acc = fmaf(v, sW[h], acc);
    }
    acc += bd2[0];
    Apred[(size_t)(i0 + ti) * N + (j0 + tj)] = 1.f / (1.f + expf(-acc));
}

// ---------------------------------------------------------------------------
extern "C" void kernel_launch(void* const* d_in, const int* in_sizes, int n_in,
                              void* d_out, int out_size, void* d_ws, size_t ws_size,
                              hipStream_t stream)
{
    (void)in_sizes; (void)n_in; (void)out_size; (void)ws_size;

    const float* A_hat = (const float*)d_in[0];   // [N,N]
    const float* X     = (const float*)d_in[1];   // [N,D]
    const float* eps   = (const float*)d_in[2];   // [N,L]
    const float* W1    = (const float*)d_in[3];   // [H,D]
    const float* b1    = (const float*)d_in[4];   // [H]
    const float* Wmu   = (const float*)d_in[5];   // [L,H]
    const float* bmu   = (const float*)d_in[6];   // [L]
    const float* Wlv   = (const float*)d_in[7];   // [L,H]
    const float* blv   = (const float*)d_in[8];   // [L]
    const float* Wd1   = (const float*)d_in[9];   // [2L,2L]
    const float* bd1   = (const float*)d_in[10];  // [2L]
    const float* Wd2   = (const float*)d_in[11];  // [1,2L]
    const float* bd2   = (const float*)d_in[12];  // [1]

    const int N = 2048, D = 512, H = 256, L = 16;

    // Workspace layout (floats). AH aliases T1: T1 is dead after GEMM2.
    float* ws = (float*)d_ws;
    float* T1 = ws;                         // [N,D]  = A_hat @ X
    float* T2 = T1 + (size_t)N * D;         // [N,H]  = leaky(T1 @ W1^T + b1)
    float* Pm = T2 + (size_t)N * H;         // [N,2L]
    float* Qm = Pm + (size_t)N * 2 * L;     // [N,2L]
    float* AH = T1;                         // [N,H]  = A_hat @ T2 (reuses T1)

    float* Apred = (float*)d_out;                 // [N,N]
    float* mu    = Apred + (size_t)N * N;         // [N,L]
    float* lv    = mu + (size_t)N * L;            // [N,L]

    const dim3 gblk(128);

    // GEMM 1: T1 = A_hat @ X           (M=2048, N=512, K=2048)
    gemm_wmma_f32<0, false><<<dim3(D / 64, N / 64), gblk, 0, stream>>>(
        A_hat, X, nullptr, T1, N, D, N, /*lda=*/N, /*ldb=*/D, /*ldc=*/D);

    // GEMM 2: T2 = leaky(T1 @ W1^T + b1)  (M=2048, N=256, K=512, B transposed)
    gemm_wmma_f32<1, true><<<dim3(H / 64, N / 64), gblk, 0, stream>>>(
        T1, W1, b1, T2, N, H, D, /*lda=*/D, /*ldb=*/D, /*ldc=*/H);

    // GEMM 3: AH = A_hat @ T2          (M=2048, N=256, K=2048)
    gemm_wmma_f32<0, false><<<dim3(H / 64, N / 64), gblk, 0, stream>>>(
        A_hat, T2, nullptr, AH, N, H, N, /*lda=*/N, /*ldb=*/H, /*ldc=*/H);

    // mu / logvar / Z / P / Q (writes mu, lv into d_out)
    encode_tail<<<dim3(N), dim3(64), 0, stream>>>(
        AH, eps, Wmu, bmu, Wlv, blv, Wd1, mu, lv, Pm, Qm, H, L);

    // Decoder: A_pred
    decode_tile<<<dim3(N / 16, N / 16), dim3(256), 0, stream>>>(
        Pm, Qm, bd1, Wd2, bd2, Apred, N);
}